// NCC_61143154425979
// MI455X (gfx1250) — compile-verified
//
#include <hip/hip_runtime.h>

typedef float v2f __attribute__((ext_vector_type(2)));
typedef float v8f __attribute__((ext_vector_type(8)));

#define NVOX   160
#define WIN    11
#define PADK   5
#define TILE   16
#define PADDED 26            // TILE + 2*PADK
#define LROW   28            // padded LDS row stride (floats), even, > PADDED+1
#define LROWS  32            // padded LDS row count (covers both WMMA rowgroups)
#define VOLW   1331.0f
#define EPSW   0.0001f

#define GLOBAL_AS __attribute__((address_space(1)))
#define LDS_AS    __attribute__((address_space(3)))

#if __has_builtin(__builtin_amdgcn_global_load_async_to_lds_b32) && \
    __has_builtin(__builtin_amdgcn_s_wait_asynccnt)
#define HAVE_ASYNC 1
#else
#define HAVE_ASYNC 0
#endif

// One wave32 per workgroup. Workgroup owns a 16x16 (x,y) tile and a 16-slice
// z chunk; slides an 11-deep window along z using an LDS ring buffer.
// Separable 11-wide box sums are banded-matrix matmuls on V_WMMA_F32_16X16X4_F32.
__global__ __launch_bounds__(32)
void ncc_main(const float* __restrict__ I, const float* __restrict__ J,
              float* __restrict__ partial)
{
    __shared__ float lsI[LROWS * LROW];          //  3584 B, zero-padded
    __shared__ float lsJ[LROWS * LROW];          //  3584 B, zero-padded
    __shared__ float rbuf[5][TILE * 32];         // 10240 B, transposed: [xo][k]
    __shared__ float ring[5 * WIN * 256];        // 56320 B, lane-major v8f slices

    const int lane = threadIdx.x;
    const int half = lane >> 4;     // 0: lanes 0-15, 1: lanes 16-31
    const int lm   = lane & 15;

    int wg = blockIdx.x;
    const int bx = wg % 10;
    const int by = (wg / 10) % 10;
    const int bz = (wg / 100) % 10;
    const int bb = wg / 1000;
    const int x0 = bx * TILE, y0 = by * TILE, zb = bz * TILE;

    // Zero LDS input tiles once: halo/padding positions are never overwritten,
    // so all operand loads below can be unconditional.
    for (int idx = lane; idx < LROWS * LROW; idx += 32) {
        lsI[idx] = 0.0f;
        lsJ[idx] = 0.0f;
    }

    // Banded ones matrix chunks for WMMA B operand (4x16 f32):
    //   VGPR j : lanes 0-15 -> K = j, lanes 16-31 -> K = j+2 ; N = lane&15
    // band(K, N) = 1  iff  N <= K <= N+10 and K < 26
    v2f bB[7];
#pragma unroll
    for (int c = 0; c < 7; ++c) {
#pragma unroll
        for (int j = 0; j < 2; ++j) {
            int K = c * 4 + j + 2 * half;
            bB[c][j] = (K < PADDED && lm <= K && K <= lm + (WIN - 1)) ? 1.0f : 0.0f;
        }
    }

    const v8f zero8 = {0, 0, 0, 0, 0, 0, 0, 0};
    v8f zs0 = zero8, zs1 = zero8, zs2 = zero8, zs3 = zero8, zs4 = zero8;
    float acc = 0.0f;

    for (int s = 0; s < PADDED; ++s) {
        const int z = zb - PADK + s;

        // All DS reads of the previous slice must complete before new data
        // lands in lsI/lsJ (async DMA writes are not ordered against DS ops).
        asm volatile("s_wait_dscnt 0x0" ::: "memory");

        // ---- fill padded (x,y) slice of I and J ----
#if HAVE_ASYNC
        if (z >= 0 && z < NVOX) {
            for (int idx = lane; idx < PADDED * PADDED; idx += 32) {
                int py = idx / PADDED, px = idx - py * PADDED;
                int gx = x0 + px - PADK, gy = y0 + py - PADK;
                if (gx >= 0 && gx < NVOX && gy >= 0 && gy < NVOX) {
                    size_t off = (((size_t)(bb * NVOX + z) * NVOX + gy) * NVOX) + gx;
                    __builtin_amdgcn_global_load_async_to_lds_b32(
                        (GLOBAL_AS int*)(I + off), (LDS_AS int*)&lsI[py * LROW + px], 0, 0);
                    __builtin_amdgcn_global_load_async_to_lds_b32(
                        (GLOBAL_AS int*)(J + off), (LDS_AS int*)&lsJ[py * LROW + px], 0, 0);
                }
            }
            __builtin_amdgcn_s_wait_asynccnt(0);
            asm volatile("" ::: "memory");
        } else {
            for (int idx = lane; idx < PADDED * PADDED; idx += 32) {
                int py = idx / PADDED, px = idx - py * PADDED;
                lsI[py * LROW + px] = 0.0f;
                lsJ[py * LROW + px] = 0.0f;
            }
        }
#else
        for (int idx = lane; idx < PADDED * PADDED; idx += 32) {
            int py = idx / PADDED, px = idx - py * PADDED;
            int gx = x0 + px - PADK, gy = y0 + py - PADK;
            bool ok = (z >= 0) & (z < NVOX) & (gx >= 0) & (gx < NVOX) & (gy >= 0) & (gy < NVOX);
            size_t off = (((size_t)(bb * NVOX + (ok ? z : 0)) * NVOX + (ok ? gy : 0)) * NVOX) + (ok ? gx : 0);
            lsI[py * LROW + px] = ok ? I[off] : 0.0f;
            lsJ[py * LROW + px] = ok ? J[off] : 0.0f;
        }
#endif

        // ---- X pass: rowsum[y][xo] = sum_k in[y][k] * band[k][xo] (WMMA) ----
        // A 16x4 f32: lane (half,lm): M = lm (+16*rg), K = c*4 + 2*half + j.
        // Padded rows/cols read exact zeros, so everything is unconditional.
#pragma unroll
        for (int rg = 0; rg < 2; ++rg) {
            v8f d0 = zero8, d1 = zero8, d2 = zero8, d3 = zero8, d4 = zero8;
            const int row = rg * 16 + lm;
#pragma unroll
            for (int c = 0; c < 7; ++c) {
                const int kb = c * 4 + 2 * half;           // even -> 8B aligned
                v2f aI = *(const v2f*)&lsI[row * LROW + kb];   // ds_load_b64
                v2f aJ = *(const v2f*)&lsJ[row * LROW + kb];   // ds_load_b64
                v2f aII = aI * aI;
                v2f aJJ = aJ * aJ;
                v2f aIJ = aI * aJ;
                d0 = __builtin_amdgcn_wmma_f32_16x16x4_f32(false, aI,  false, bB[c], (short)0, d0, false, false);
                d1 = __builtin_amdgcn_wmma_f32_16x16x4_f32(false, aJ,  false, bB[c], (short)0, d1, false, false);
                d2 = __builtin_amdgcn_wmma_f32_16x16x4_f32(false, aII, false, bB[c], (short)0, d2, false, false);
                d3 = __builtin_amdgcn_wmma_f32_16x16x4_f32(false, aJJ, false, bB[c], (short)0, d3, false, false);
                d4 = __builtin_amdgcn_wmma_f32_16x16x4_f32(false, aIJ, false, bB[c], (short)0, d4, false, false);
            }
            // Transposed store rbuf[f][N*32 + M], M = r + 8*half + 16*rg:
            // 8 consecutive floats per lane -> vector DS store. Rows M>=26 of
            // rowgroup 1 are exact zeros (zero A rows), safe to store.
            const int rb = lm * 32 + 8 * half + 16 * rg;
            *(v8f*)&rbuf[0][rb] = d0;
            *(v8f*)&rbuf[1][rb] = d1;
            *(v8f*)&rbuf[2][rb] = d2;
            *(v8f*)&rbuf[3][rb] = d3;
            *(v8f*)&rbuf[4][rb] = d4;
        }

        // ---- Y pass + sliding z window ----
        const int slot = s % WIN;
        auto ypass = [&](const float* fb, v8f& zsum, int fidx) {
            v8f s2 = zero8;
#pragma unroll
            for (int c = 0; c < 7; ++c) {
                const int kb = c * 4 + 2 * half;           // A2[m=xo][k=y]
                v2f a = *(const v2f*)&fb[lm * 32 + kb];     // ds_load_b64
                s2 = __builtin_amdgcn_wmma_f32_16x16x4_f32(false, a, false, bB[c], (short)0, s2, false, false);
            }
            v8f* rp = (v8f*)&ring[(fidx * WIN + slot) * 256 + lane * 8];
            v8f old = (s >= WIN) ? *rp : zero8;             // slice leaving window
            zsum += s2 - old;
            *rp = s2;
        };
        ypass(rbuf[0], zs0, 0);
        ypass(rbuf[1], zs1, 1);
        ypass(rbuf[2], zs2, 2);
        ypass(rbuf[3], zs3, 3);
        ypass(rbuf[4], zs4, 4);

        // ---- emit cc for output slice zo = zb + (s - 10) ----
        if (s >= WIN - 1) {
            const float inv_vol = 1.0f / VOLW;
#pragma unroll
            for (int r = 0; r < 8; ++r) {
                float Is = zs0[r], Js = zs1[r], I2 = zs2[r], J2 = zs3[r], IJ = zs4[r];
                float cross = fmaxf(IJ - Is * Js * inv_vol, EPSW);
                float iv    = fmaxf(I2 - Is * Is * inv_vol, EPSW);
                float jv    = fmaxf(J2 - Js * Js * inv_vol, EPSW);
                acc += cross / sqrtf(iv * jv);
            }
        }
    }

    // wave reduction (wave32)
#pragma unroll
    for (int off = 16; off > 0; off >>= 1)
        acc += __shfl_xor(acc, off, 32);
    if (lane == 0) partial[blockIdx.x] = acc;
}

__global__ __launch_bounds__(256)
void ncc_finalize(const float* __restrict__ partial, float* __restrict__ out,
                  int nparts, float inv_count)
{
    __shared__ float sm[256];
    float t = 0.0f;
    for (int i = threadIdx.x; i < nparts; i += 256) t += partial[i];
    sm[threadIdx.x] = t;
    for (int w = 128; w > 0; w >>= 1) {
        __syncthreads();
        if ((int)threadIdx.x < w) sm[threadIdx.x] += sm[threadIdx.x + w];
    }
    __syncthreads();
    if (threadIdx.x == 0) out[0] = 1.0f - sm[0] * inv_count;
}

extern "C" void kernel_launch(void* const* d_in, const int* in_sizes, int n_in,
                              void* d_out, int out_size, void* d_ws, size_t ws_size,
                              hipStream_t stream) {
    const float* I = (const float*)d_in[0];
    const float* J = (const float*)d_in[1];
    float* partial = (float*)d_ws;            // 2000 floats
    (void)in_sizes; (void)n_in; (void)out_size; (void)ws_size;

    ncc_main<<<2000, 32, 0, stream>>>(I, J, partial);
    ncc_finalize<<<1, 256, 0, stream>>>(partial, (float*)d_out,
                                        2000, 1.0f / 8192000.0f);
}